// BicliqueAttentionLayer_36679020707948
// MI455X (gfx1250) — compile-verified
//
#include <hip/hip_runtime.h>

typedef __attribute__((ext_vector_type(2))) float v2f;
typedef __attribute__((ext_vector_type(8))) float v8f;

#define IN_DIM  128
#define OUT_DIM 128
#define HEADS   4
#define HDIM    32

// ---------------------------------------------------------------------------
// init: zero output accumulator, -inf segment max, zero denominators
// ---------------------------------------------------------------------------
__global__ void k_init(float* __restrict__ out, float* __restrict__ segmax,
                       float* __restrict__ denom, int n) {
  int idx = blockIdx.x * blockDim.x + threadIdx.x;
  if (idx < n * OUT_DIM) out[idx] = 0.0f;
  if (idx < n * HEADS) {
    segmax[idx] = -__builtin_inff();
    denom[idx]  = 0.0f;
  }
}

// ---------------------------------------------------------------------------
// Projection: h = feat @ (diag(mask) * W) via V_WMMA_F32_16X16X4_F32.
// One 16x16 output tile per wave32; 32 WMMA steps over K=128.
// W is pre-scaled by mask and staged in LDS (64 KB of the 320 KB WGP LDS).
// Fragment layouts per CDNA5 ISA 7.12.2:
//   A 16x4  : lanes 0-15 hold M=lane, K={0,1}; lanes 16-31 hold M=lane-16, K={2,3}
//   B 4x16  : VGPR0 = rows K=0 (lanes 0-15) / K=2 (lanes 16-31); VGPR1 = K=1/K=3
//   C/D 16x16: VGPR j, lane l -> M = j + 8*(l>>4), N = l&15
// ---------------------------------------------------------------------------
__global__ void __launch_bounds__(128) k_proj_wmma(
    const float* __restrict__ feat, const float* __restrict__ mask,
    const float* __restrict__ W, float* __restrict__ hbuf,
    int n, int nTiles) {
  __shared__ float ldsW[IN_DIM * OUT_DIM];

  // Stage mask-scaled W: (feat*mask)@W == feat @ (diag(mask)W)
  for (int i = threadIdx.x; i < IN_DIM * OUT_DIM; i += 128) {
    int k = i >> 7;             // row index = input-feature index
    ldsW[i] = W[i] * mask[k];
  }
  __syncthreads();

  const int wid  = threadIdx.x >> 5;
  const int lane = threadIdx.x & 31;
  const int tile = blockIdx.x * 4 + wid;
  if (tile >= nTiles) return;   // wave-uniform: EXEC stays all-ones inside
  const int rowTile = tile >> 3;        // OUT_DIM/16 == 8 col tiles
  const int colTile = tile & 7;

  const int m     = lane & 15;
  const int khalf = (lane >> 4) << 1;   // 0 or 2
  int row = rowTile * 16 + m;
  if (row >= n) row = n - 1;            // clamp (keeps EXEC full for WMMA)
  const int col = colTile * 16 + m;

  const float* __restrict__ fp = feat + (size_t)row * IN_DIM + khalf;
  const float* __restrict__ w0 = ldsW + (size_t)khalf * OUT_DIM + col;

  v8f acc = {};
  #pragma unroll
  for (int k0 = 0; k0 < IN_DIM; k0 += 4) {
    v2f a, b;
    a.x = fp[k0];                       // feat[row][k0+khalf]   (contiguous pair)
    a.y = fp[k0 + 1];                   // feat[row][k0+khalf+1]
    b.x = w0[(size_t)k0 * OUT_DIM];             // W'[k0+khalf  ][col]
    b.y = w0[(size_t)(k0 + 1) * OUT_DIM];       // W'[k0+khalf+1][col]
    acc = __builtin_amdgcn_wmma_f32_16x16x4_f32(
        /*neg_a=*/false, a, /*neg_b=*/false, b,
        /*c_mod=*/(short)0, acc, /*reuse_a=*/false, /*reuse_b=*/false);
  }

  const int mBase = rowTile * 16 + 8 * (lane >> 4);
  const int cN    = colTile * 16 + (lane & 15);
  #pragma unroll
  for (int j = 0; j < 8; ++j) {
    int r = mBase + j;
    if (r < n) hbuf[(size_t)r * OUT_DIM + cN] = acc[j];
  }
}

// ---------------------------------------------------------------------------
// Per-(node, head) attention logits: leaky_relu(h[n,h,:] . attn[h,:], 0.01)
// ---------------------------------------------------------------------------
__global__ void k_score(const float* __restrict__ hbuf,
                        const float* __restrict__ attn,
                        float* __restrict__ score, int n) {
  int idx = blockIdx.x * blockDim.x + threadIdx.x;
  if (idx >= n * HEADS) return;
  int node = idx >> 2;
  int h    = idx & 3;
  const float* hp = hbuf + (size_t)node * OUT_DIM + h * HDIM;
  const float* ap = attn + h * HDIM;
  float s = 0.0f;
  #pragma unroll
  for (int d = 0; d < HDIM; ++d) s += hp[d] * ap[d];
  score[idx] = (s > 0.0f) ? s : 0.01f * s;
}

// ---------------------------------------------------------------------------
// Deterministic float atomic-max via signed/unsigned int ordering trick
// (bit pattern of positive floats orders as signed int; negatives order
// inversely as unsigned). segmax initialized to -inf (0xFF800000).
// ---------------------------------------------------------------------------
__device__ __forceinline__ void atomic_max_f32(float* addr, float v) {
  if (v >= 0.0f)
    atomicMax((int*)addr, __float_as_int(v));
  else
    atomicMin((unsigned int*)addr, __float_as_uint(v));
}

__global__ void k_segmax(const int* __restrict__ src, const int* __restrict__ dst,
                         const float* __restrict__ score,
                         float* __restrict__ segmax, int e) {
  int idx = blockIdx.x * blockDim.x + threadIdx.x;
  if (idx >= e * HEADS) return;
  int ed = idx >> 2;
  int h  = idx & 3;
  float v = score[src[ed] * HEADS + h];
  atomic_max_f32(&segmax[dst[ed] * HEADS + h], v);
}

__global__ void k_denom(const int* __restrict__ src, const int* __restrict__ dst,
                        const float* __restrict__ score,
                        const float* __restrict__ segmax,
                        float* __restrict__ denom, int e) {
  int idx = blockIdx.x * blockDim.x + threadIdx.x;
  if (idx >= e * HEADS) return;
  int ed = idx >> 2;
  int h  = idx & 3;
  int d  = dst[ed] * HEADS + h;
  float ex = __expf(score[src[ed] * HEADS + h] - segmax[d]);
  atomicAdd(&denom[d], ex);
}

// ---------------------------------------------------------------------------
// Weighted message scatter: one wave32 per edge, 4 columns per lane
// (column c = j*32 + lane -> head j, so alpha is lane-uniform per j).
// Gathers from L2-resident h, scatters with non-returning atomic fadd.
// ---------------------------------------------------------------------------
__global__ void __launch_bounds__(256) k_scatter(
    const int* __restrict__ src, const int* __restrict__ dst,
    const float* __restrict__ hbuf, const float* __restrict__ score,
    const float* __restrict__ segmax, const float* __restrict__ denom,
    float* __restrict__ out, int e) {
  const int wid  = threadIdx.x >> 5;
  const int lane = threadIdx.x & 31;
  const int ed   = blockIdx.x * 8 + wid;
  if (ed >= e) return;
  const int s = src[ed];
  const int d = dst[ed];
  const float* __restrict__ hs = hbuf + (size_t)s * OUT_DIM;
  float* __restrict__ od = out + (size_t)d * OUT_DIM;
  __builtin_prefetch(hs + lane, 0, 0);   // global_prefetch_b8 on the src row
  #pragma unroll
  for (int j = 0; j < HEADS; ++j) {
    int sh = s * HEADS + j;
    int dh = d * HEADS + j;
    float alpha = __expf(score[sh] - segmax[dh]) / denom[dh];
    atomicAdd(&od[j * 32 + lane], hs[j * 32 + lane] * alpha);
  }
}

// ---------------------------------------------------------------------------
// Final in-place relu on accumulated output
// ---------------------------------------------------------------------------
__global__ void k_relu(float* __restrict__ out, int total) {
  int idx = blockIdx.x * blockDim.x + threadIdx.x;
  if (idx < total) out[idx] = fmaxf(out[idx], 0.0f);
}

// ---------------------------------------------------------------------------
extern "C" void kernel_launch(void* const* d_in, const int* in_sizes, int n_in,
                              void* d_out, int out_size, void* d_ws, size_t ws_size,
                              hipStream_t stream) {
  const float* feat = (const float*)d_in[0];   // [N,128]
  const float* mask = (const float*)d_in[1];   // [128]
  const float* W    = (const float*)d_in[2];   // [128,128]
  const float* attn = (const float*)d_in[3];   // [4,32]
  const int*   src  = (const int*)d_in[4];     // [E]
  const int*   dst  = (const int*)d_in[5];     // [E]
  float* out = (float*)d_out;                  // [N,128]

  const int n = in_sizes[0] / IN_DIM;
  const int e = in_sizes[4];

  // workspace layout (floats): hbuf[N*128] | score[N*4] | segmax[N*4] | denom[N*4]
  float* hbuf   = (float*)d_ws;
  float* score  = hbuf   + (size_t)n * OUT_DIM;
  float* segmax = score  + (size_t)n * HEADS;
  float* denom  = segmax + (size_t)n * HEADS;

  const int total = n * OUT_DIM;

  k_init<<<(total + 255) / 256, 256, 0, stream>>>(out, segmax, denom, n);

  const int rowTiles = (n + 15) / 16;
  const int nTiles   = rowTiles * (OUT_DIM / 16);
  k_proj_wmma<<<(nTiles + 3) / 4, 128, 0, stream>>>(feat, mask, W, hbuf, n, nTiles);

  k_score<<<(n * HEADS + 255) / 256, 256, 0, stream>>>(hbuf, attn, score, n);

  k_segmax<<<(e * HEADS + 255) / 256, 256, 0, stream>>>(src, dst, score, segmax, e);
  k_denom <<<(e * HEADS + 255) / 256, 256, 0, stream>>>(src, dst, score, segmax, denom, e);

  k_scatter<<<(e + 7) / 8, 256, 0, stream>>>(src, dst, hbuf, score, segmax, denom, out, e);

  k_relu<<<(total + 255) / 256, 256, 0, stream>>>(out, total);
}